// FluidSimulator_88742614270225
// MI455X (gfx1250) — compile-verified
//
#include <hip/hip_runtime.h>
#include <stdint.h>

#define N   192
#define NN  (N * N)
#define N3  (N * N * N)

// Physics constants (match reference)
#define DT_C     0.01f
#define H_C      0.1f
#define VISC_C   0.01f
#define GRAV_C   9.81f
#define GZ_C     (-GRAV_C * DT_C)
#define DT_H_C   (DT_C / H_C)                // 0.1
#define ALPHA_C  (VISC_C * DT_C / (H_C*H_C)) // 0.01
#define INVD_C   (1.0f / (1.0f + 6.0f * ALPHA_C))
#define INV2H_C  (1.0f / (2.0f * H_C))       // 5.0

// Tile geometry for LDS-tiled Jacobi sweeps
#define TX 8
#define TY 8
#define TZ 32
#define LX (TX + 2)
#define LY (TY + 2)
#define LZ (TZ + 2)
#define LTOT (LX * LY * LZ)   // 3400 floats = 13.6 KB

typedef uint32_t u32x4 __attribute__((ext_vector_type(4)));
typedef int      i32x4 __attribute__((ext_vector_type(4)));
typedef int      i32x8 __attribute__((ext_vector_type(8)));

// ---------------------------------------------------------------------------
// Per-lane CDNA5 async DMA tile fill (ASYNCcnt path). Low 32 bits of a generic
// LDS pointer are the wave-relative LDS byte offset (ISA 10.2 aperture map),
// matching the async op's VDST semantics.
// ---------------------------------------------------------------------------
__device__ __forceinline__ void load_tile_async(const float* __restrict__ base,
                                                float* tile,
                                                int x0, int y0, int z0, int tid) {
  for (int e = tid; e < LTOT; e += 256) {
    int lz = e % LZ;
    int r  = e / LZ;
    int ly = r % LY;
    int lx = r / LY;
    int gx = min(x0 - 1 + lx, N - 1);
    int gy = min(y0 - 1 + ly, N - 1);
    int gz = min(z0 - 1 + lz, N - 1);
    const float* gp = base + ((gx * N + gy) * N + gz);
    uint32_t lds_off = (uint32_t)(uintptr_t)(&tile[e]);
    asm volatile("global_load_async_to_lds_b32 %0, %1, off"
                 :
                 : "v"(lds_off), "v"((uint64_t)(uintptr_t)gp)
                 : "memory");
  }
  asm volatile("s_wait_asynccnt 0x0" ::: "memory");
  __syncthreads();
}

// ---------------------------------------------------------------------------
// TDM tile fill: one tensor_load_to_lds per block (TENSORcnt path).
// D# per ISA 8.3-8.6: group0 = count/lds_addr/global_addr/type=2,
// group1 = data_size=4B, tile 34x10x10, strides 192 / 36864,
// groups 2-3 carry dim2 (x). tensor_dim* = extent remaining from tile start
// so edge tiles get hardware zero-fill instead of OOB reads.
// ---------------------------------------------------------------------------
#if __has_builtin(__builtin_amdgcn_tensor_load_to_lds)
#define HAVE_TDM 1
__device__ __forceinline__ void load_tile_tdm(const float* __restrict__ base,
                                              float* tile,
                                              int x0, int y0, int z0, int tid) {
  if (tid == 0) {
    const int gx0 = x0 - 1, gy0 = y0 - 1, gz0 = z0 - 1;
    uint64_t ga = (uint64_t)(uintptr_t)(base + ((size_t)gx0 * N + gy0) * N + gz0);
    uint32_t lds_base = (uint32_t)(uintptr_t)tile;

    u32x4 g0;
    g0[0] = 1u;                                   // count=1, user mode
    g0[1] = lds_base;                             // lds_addr (bytes)
    g0[2] = (uint32_t)ga;                         // global_addr[31:0]
    g0[3] = (uint32_t)((ga >> 32) & 0x01FFFFFFu)  // global_addr[56:32]
          | 0x80000000u;                          // type=2 ("image")

    uint32_t td0 = (uint32_t)(N - gz0);           // remaining extent, dim0 (z)
    uint32_t td1 = (uint32_t)(N - gy0);           // dim1 (y)
    uint32_t td2 = (uint32_t)(N - gx0);           // dim2 (x)
    i32x8 g1;
    g1[0] = (int)0x00020000u;                     // wg_mask=0, data_size=2 (4B)
    g1[1] = (int)(td0 << 16);                     // abar=0 | tensor_dim0 lo16
    g1[2] = (int)((td0 >> 16) | (td1 << 16));     // tensor_dim0 hi | dim1 lo
    g1[3] = (int)((td1 >> 16) | ((uint32_t)LZ << 16)); // dim1 hi | tile_dim0=34
    g1[4] = (int)((uint32_t)LY | ((uint32_t)LX << 16)); // tile_dim1=10|tile_dim2=10
    g1[5] = (int)N;                               // tensor_dim0_stride = 192
    g1[6] = (int)((uint32_t)NN << 16);            // stride0 hi=0 | stride1 lo16 (36864)
    g1[7] = (int)((uint32_t)NN >> 16);            // stride1 hi32 (=0)

    i32x4 g2;
    g2[0] = (int)td2;                             // tensor_dim2
    g2[1] = 1;                                    // tensor_dim3 (unused)
    g2[2] = 0;                                    // tensor_dim2_stride lo (unused)
    g2[3] = 0;                                    // stride hi | tile_dim3=0

    i32x4 g3;
    g3[0] = 0;                                    // tensor_dim3_stride
    g3[1] = (int)(1u << 16);                      // tensor_dim4=1 (unused)
    g3[2] = 0;                                    // tile_dim4=0
    g3[3] = 0;

#if defined(__clang_major__) && (__clang_major__ >= 23)
    i32x8 gz8 = {0, 0, 0, 0, 0, 0, 0, 0};
    __builtin_amdgcn_tensor_load_to_lds(g0, g1, g2, g3, gz8, 0);
#else
    __builtin_amdgcn_tensor_load_to_lds(g0, g1, g2, g3, 0);
#endif
  }
  __builtin_amdgcn_s_wait_tensorcnt(0);
  __syncthreads();
}
#else
#define HAVE_TDM 0
#endif

// ---------------------------------------------------------------------------
// 1) gravity + semi-Lagrangian advection (fused, 3D launch: no int division)
// ---------------------------------------------------------------------------
__global__ __launch_bounds__(192) void advect_kernel(
    const float* __restrict__ vin, float* __restrict__ vout) {
  const int z = threadIdx.x;
  const int y = blockIdx.x;
  const int x = blockIdx.y;
  const int idx = (x * N + y) * N + z;

  float v0 = vin[idx];
  float v1 = vin[idx + N3];
  float v2 = vin[idx + 2 * N3] + GZ_C;

  float px = (float)x - v0 * DT_H_C;
  float py = (float)y - v1 * DT_H_C;
  float pz = (float)z - v2 * DT_H_C;
  const float hi = (float)(N - 2);
  px = fminf(fmaxf(px, 0.0f), hi);
  py = fminf(fmaxf(py, 0.0f), hi);
  pz = fminf(fmaxf(pz, 0.0f), hi);

  int ix = (int)px, iy = (int)py, iz = (int)pz;   // floor (non-negative)
  float wx1 = px - (float)ix, wx0 = 1.0f - wx1;
  float wy1 = py - (float)iy, wy0 = 1.0f - wy1;
  float wz1 = pz - (float)iz, wz0 = 1.0f - wz1;

  int b = (ix * N + iy) * N + iz;
  int o000 = b,          o001 = b + 1;
  int o010 = b + N,      o011 = b + N + 1;
  int o100 = b + NN,     o101 = b + NN + 1;
  int o110 = b + NN + N, o111 = b + NN + N + 1;

  float w000 = wx0 * wy0 * wz0, w001 = wx0 * wy0 * wz1;
  float w010 = wx0 * wy1 * wz0, w011 = wx0 * wy1 * wz1;
  float w100 = wx1 * wy0 * wz0, w101 = wx1 * wy0 * wz1;
  float w110 = wx1 * wy1 * wz0, w111 = wx1 * wy1 * wz1;

#pragma unroll
  for (int c = 0; c < 3; ++c) {
    const float* f = vin + c * N3;
    float r = f[o000] * w000 + f[o001] * w001 + f[o010] * w010 + f[o011] * w011
            + f[o100] * w100 + f[o101] * w101 + f[o110] * w110 + f[o111] * w111;
    if (c == 2) r += GZ_C;
    vout[idx + c * N3] = r;
  }
}

// ---------------------------------------------------------------------------
// 2) one LDS-tiled Jacobi diffusion sweep for one velocity component
//    (interior only; boundaries pre-seeded in both ping-pong buffers).
//    blockIdx.z folds {component, x-tile}.
// ---------------------------------------------------------------------------
__global__ __launch_bounds__(256) void diffuse_tile_kernel(
    const float* __restrict__ vin, float* __restrict__ vout) {
  __shared__ float tile[LTOT];
  const int c  = blockIdx.z / ((N - 2 + TX - 1) / TX);
  const int xt = blockIdx.z % ((N - 2 + TX - 1) / TX);
  const int x0 = 1 + xt * TX;
  const int y0 = 1 + blockIdx.y * TY;
  const int z0 = 1 + blockIdx.x * TZ;
  const int tid = threadIdx.y * 32 + threadIdx.x;   // blockDim = (32,8)

  const float* cin  = vin  + (size_t)c * N3;
  float*       cout = vout + (size_t)c * N3;

  load_tile_async(cin, tile, x0, y0, z0, tid);

  const int lz = threadIdx.x + 1;
  const int ly = threadIdx.y + 1;
  const int gz = z0 + (int)threadIdx.x;
  const int gy = y0 + (int)threadIdx.y;
  const bool okyz = (gz <= N - 2) && (gy <= N - 2);

#pragma unroll
  for (int lx = 1; lx <= TX; ++lx) {
    int gx = x0 + lx - 1;
    if (okyz && gx <= N - 2) {
      int li = (lx * LY + ly) * LZ + lz;
      float s = tile[li + LY * LZ] + tile[li - LY * LZ]
              + tile[li + LZ]      + tile[li - LZ]
              + tile[li + 1]       + tile[li - 1];
      cout[(gx * N + gy) * N + gz] = (tile[li] + ALPHA_C * s) * INVD_C;
    }
  }
}

// ---------------------------------------------------------------------------
// 3) divergence + first pressure sweep fused:
//    dv = div(vel); pA = dv/6 interior (Jacobi sweep 1 from p=0); pB = 0.
// ---------------------------------------------------------------------------
__global__ __launch_bounds__(192) void divpinit_kernel(
    const float* __restrict__ vel, float* __restrict__ dv,
    float* __restrict__ pA, float* __restrict__ pB) {
  const int z = threadIdx.x;
  const int y = blockIdx.x;
  const int x = blockIdx.y;
  const int idx = (x * N + y) * N + z;
  pB[idx] = 0.0f;
  if (x == 0 || x == N - 1 || y == 0 || y == N - 1 || z == 0 || z == N - 1) {
    dv[idx] = 0.0f;
    pA[idx] = 0.0f;
    return;
  }
  float d = (vel[idx + NN] - vel[idx - NN])
          + (vel[N3 + idx + N] - vel[N3 + idx - N])
          + (vel[2 * N3 + idx + 1] - vel[2 * N3 + idx - 1]);
  d *= INV2H_C;
  dv[idx] = d;
  pA[idx] = d * (1.0f / 6.0f);
}

// ---------------------------------------------------------------------------
// 4) pressure Jacobi sweep, LDS-tiled; tile fetched by the Tensor Data Mover
//    (one DMA per block) when available, else per-lane async DMA.
// ---------------------------------------------------------------------------
__global__ __launch_bounds__(256) void pjacobi_kernel(
    const float* __restrict__ pin, const float* __restrict__ dv,
    float* __restrict__ pout) {
  __shared__ float tile[LTOT];
  const int x0 = 1 + blockIdx.z * TX;
  const int y0 = 1 + blockIdx.y * TY;
  const int z0 = 1 + blockIdx.x * TZ;
  const int tid = threadIdx.y * 32 + threadIdx.x;   // blockDim = (32,8)

#if HAVE_TDM
  load_tile_tdm(pin, tile, x0, y0, z0, tid);
#else
  load_tile_async(pin, tile, x0, y0, z0, tid);
#endif

  const int lz = threadIdx.x + 1;
  const int ly = threadIdx.y + 1;
  const int gz = z0 + (int)threadIdx.x;
  const int gy = y0 + (int)threadIdx.y;
  const bool okyz = (gz <= N - 2) && (gy <= N - 2);

#pragma unroll
  for (int lx = 1; lx <= TX; ++lx) {
    int gx = x0 + lx - 1;
    if (okyz && gx <= N - 2) {
      int li = (lx * LY + ly) * LZ + lz;
      float s = tile[li + LY * LZ] + tile[li - LY * LZ]
              + tile[li + LZ]      + tile[li - LZ]
              + tile[li + 1]       + tile[li - 1];
      int g = (gx * N + gy) * N + gz;
      pout[g] = (dv[g] + s) * (1.0f / 6.0f);
    }
  }
}

// ---------------------------------------------------------------------------
// 5) subtract pressure gradient (in place on the velocity field)
// ---------------------------------------------------------------------------
__global__ __launch_bounds__(192) void grad_kernel(
    const float* __restrict__ p, float* __restrict__ vel) {
  const int z = threadIdx.x;
  const int y = blockIdx.x;
  const int x = blockIdx.y;
  if (x == 0 || x == N - 1 || y == 0 || y == N - 1 || z == 0 || z == N - 1)
    return;
  const int idx = (x * N + y) * N + z;
  vel[idx]          -= (p[idx + NN] - p[idx - NN]) * INV2H_C;
  vel[idx + N3]     -= (p[idx + N]  - p[idx - N])  * INV2H_C;
  vel[idx + 2 * N3] -= (p[idx + 1]  - p[idx - 1])  * INV2H_C;
}

// ---------------------------------------------------------------------------
extern "C" void kernel_launch(void* const* d_in, const int* in_sizes, int n_in,
                              void* d_out, int out_size, void* d_ws, size_t ws_size,
                              hipStream_t stream) {
  const float* vin = (const float*)d_in[0];       // [3,N,N,N]
  float* vout = (float*)d_out;                    // [3,N,N,N]

  float* velA = (float*)d_ws;                     // 3*N3
  float* dv   = velA + (size_t)3 * N3;            // N3
  float* pA   = dv + N3;                          // N3
  float* pB   = pA + N3;                          // N3  (total 6*N3 floats)

  dim3 rowGrid(N, N);       // y, x
  dim3 rowBlk(N, 1, 1);     // z row = 6 waves

  const int ZT = (N - 2 + TZ - 1) / TZ;   // 6
  const int YT = (N - 2 + TY - 1) / TY;   // 24
  const int XT = (N - 2 + TX - 1) / TX;   // 24
  dim3 tBlk(32, 8, 1);

  // gravity + advect -> velA
  advect_kernel<<<rowGrid, rowBlk, 0, stream>>>(vin, velA);

  // seed boundaries (and interior, harmlessly) of the second ping-pong buffer
  hipMemcpyAsync(vout, velA, (size_t)3 * N3 * sizeof(float),
                 hipMemcpyDeviceToDevice, stream);

  // 5 interior-only tiled diffusion sweeps, ping-pong velA <-> d_out
  dim3 dGrid(ZT, YT, XT * 3);
  diffuse_tile_kernel<<<dGrid, tBlk, 0, stream>>>(velA, vout);
  diffuse_tile_kernel<<<dGrid, tBlk, 0, stream>>>(vout, velA);
  diffuse_tile_kernel<<<dGrid, tBlk, 0, stream>>>(velA, vout);
  diffuse_tile_kernel<<<dGrid, tBlk, 0, stream>>>(vout, velA);
  diffuse_tile_kernel<<<dGrid, tBlk, 0, stream>>>(velA, vout);

  // divergence + pressure sweep 1 (p=0 => p=div/6) fused; pB zeroed
  divpinit_kernel<<<rowGrid, rowBlk, 0, stream>>>(vout, dv, pA, pB);

  // 19 more tiled pressure sweeps (TDM/async DMA), ping-pong pA <-> pB
  dim3 pGrid(ZT, YT, XT);
  float* ps = pA;
  float* pd = pB;
  for (int i = 0; i < 19; ++i) {
    pjacobi_kernel<<<pGrid, tBlk, 0, stream>>>(ps, dv, pd);
    float* tmp = ps; ps = pd; pd = tmp;
  }

  // subtract grad p (in place on d_out)
  grad_kernel<<<rowGrid, rowBlk, 0, stream>>>(ps, vout);
}